// BaseModel_27221502722520
// MI455X (gfx1250) — compile-verified
//
#include <hip/hip_runtime.h>

typedef __attribute__((ext_vector_type(2))) float v2f;
typedef __attribute__((ext_vector_type(8))) float v8f;

namespace {
constexpr int NV = 100;            // num vars (D)
constexpr int NH = 16;             // hidden dim
constexpr int NP = 2;              // params
constexpr int BT = 16;             // batch tile per workgroup (WMMA N)
constexpr int JC = 20;             // K(j) chunk size, multiple of 4, divides 100
constexpr int NCHUNK = NV / JC;    // 5
constexpr int NWAVE = 10;          // waves per workgroup
constexpr int NT = NV / NWAVE;     // 10 t's per wave
constexpr int TSTR = JC * BT + 1;  // 321 words per t-row (pad kills bank conflicts)
constexpr int ROWS_PER_WAVE = (BT * JC) / NWAVE;  // 32
}

__global__ __launch_bounds__(NWAVE * 32)
void nn_fused(const float* __restrict__ x, const float* __restrict__ M,
              const float* __restrict__ W0, const float* __restrict__ W1,
              const float* __restrict__ W2, const float* __restrict__ b0,
              const float* __restrict__ b1, const float* __restrict__ b2,
              float* __restrict__ out)
{
  __shared__ float lds_m[NV * TSTR];        // 128.4 KB: xm[t][j-in-chunk][b]
  __shared__ float lds_h[NWAVE * NH * BT];  // 10 KB per-wave relayout scratch

  const int tid  = threadIdx.x;
  const int wave = tid >> 5;
  const int lane = tid & 31;
  const int half = lane >> 4;   // WMMA half-wave
  const int lm   = lane & 15;
  const int bb0  = blockIdx.x * BT;

  v8f acc[NT];
#pragma unroll
  for (int i = 0; i < NT; ++i) acc[i] = {};

  for (int c = 0; c < NCHUNK; ++c) {
    const int jbase = c * JC;
    // ---- Phase A: stream M coalesced, mask+scale, transpose into LDS ----
    for (int rr = 0; rr < ROWS_PER_WAVE; ++rr) {
      const int r = wave * ROWS_PER_WAVE + rr;   // 320 rows = 16 b x 20 j
      const int b = r & (BT - 1);
      const int jj = r >> 4;
      const int jglob = jbase + jj;
      if (lane < NV / 4) {  // 25 lanes cover t=0..99, float4 each, 16B aligned
        const float4* src = reinterpret_cast<const float4*>(
            M + (size_t)(bb0 + b) * (NV * NV) + (size_t)jglob * NV);
        float4 v = src[lane];
        const float xv = x[(bb0 + b) * NV + jglob];
        v.x *= xv; v.y *= xv; v.z *= xv; v.w *= xv;
        const int d = jglob - 4 * lane;          // zero the j==t diagonal
        if (d == 0) v.x = 0.f;
        if (d == 1) v.y = 0.f;
        if (d == 2) v.z = 0.f;
        if (d == 3) v.w = 0.f;
        const int base = jj * BT + b;
        lds_m[(4 * lane + 0) * TSTR + base] = v.x;
        lds_m[(4 * lane + 1) * TSTR + base] = v.y;
        lds_m[(4 * lane + 2) * TSTR + base] = v.z;
        lds_m[(4 * lane + 3) * TSTR + base] = v.w;
      }
    }
    __syncthreads();
    // ---- Phase B: per-wave WMMA f32 16x16x4 over its 10 t's ----
#pragma unroll
    for (int tt = 0; tt < NT; ++tt) {
      const int t = wave * NT + tt;
      // A: lane lm holds W0[t][i=lm][k], VGPR0={K0(h0),K2(h1)}, VGPR1={K1,K3}
      const float* w0t = W0 + (size_t)t * (NH * NV) + lm * NV + jbase + 2 * half;
      // B: lane lm holds xm[k][bb=lm], rows k0+2*half, k0+1+2*half
      const float* mrow = lds_m + t * TSTR + 2 * half * BT + lm;
#pragma unroll
      for (int kk = 0; kk < JC / 4; ++kk) {
        v2f a = *reinterpret_cast<const v2f*>(w0t + 4 * kk);  // even offset -> b64
        v2f bm;
        bm.x = mrow[(4 * kk) * BT];
        bm.y = mrow[(4 * kk + 1) * BT];
        acc[tt] = __builtin_amdgcn_wmma_f32_16x16x4_f32(
            false, a, false, bm, (short)0, acc[tt], false, false);
      }
    }
    __syncthreads();
  }

  // ---- Layers 1 & 2 (per-wave, private LDS scratch) ----
  float* hs = lds_h + wave * (NH * BT);
#pragma unroll
  for (int tt = 0; tt < NT; ++tt) {
    const int t = wave * NT + tt;
    // bias + leaky ReLU on layer-0 tile; scatter to scratch as [i][b]
#pragma unroll
    for (int v = 0; v < 8; ++v) {
      const int i = v + 8 * half;               // C/D layout: row v (+8 per half)
      float hv = acc[tt][v] + b0[t * NH + i];
      hv = hv > 0.f ? hv : 0.01f * hv;
      hs[i * BT + lm] = hv;
    }
    asm volatile("s_wait_dscnt 0" ::: "memory");
    // layer 1: 16x16 @ 16x16 via 4 WMMA k-steps
    v8f a1 = {};
    const float* w1t = W1 + (size_t)t * (NH * NH) + lm * NH + 2 * half;
    const float* hrow = hs + 2 * half * BT + lm;
#pragma unroll
    for (int kk = 0; kk < NH / 4; ++kk) {
      v2f a = *reinterpret_cast<const v2f*>(w1t + 4 * kk);
      v2f bm;
      bm.x = hrow[(4 * kk) * BT];
      bm.y = hrow[(4 * kk + 1) * BT];
      a1 = __builtin_amdgcn_wmma_f32_16x16x4_f32(
          false, a, false, bm, (short)0, a1, false, false);
    }
    // layer 2 (P=2): bias+leaky then small dot; reduce across half-waves
    const float* w2t = W2 + (size_t)t * (NP * NH);
    float o0 = 0.f, o1 = 0.f;
#pragma unroll
    for (int v = 0; v < 8; ++v) {
      const int i = v + 8 * half;
      float hv = a1[v] + b1[t * NH + i];
      hv = hv > 0.f ? hv : 0.01f * hv;
      o0 += w2t[i] * hv;
      o1 += w2t[NH + i] * hv;
    }
    o0 += __shfl_xor(o0, 16, 32);
    o1 += __shfl_xor(o1, 16, 32);
    if (half == 0) {
      float2 o;
      o.x = o0 + b2[t * NP + 0];
      o.y = o1 + b2[t * NP + 1];
      *reinterpret_cast<float2*>(out + (size_t)(bb0 + lm) * (NV * NP) + t * NP) = o;
    }
  }
}

extern "C" void kernel_launch(void* const* d_in, const int* in_sizes, int n_in,
                              void* d_out, int out_size, void* d_ws, size_t ws_size,
                              hipStream_t stream) {
  const float* x  = (const float*)d_in[0];
  const float* M  = (const float*)d_in[1];
  const float* W0 = (const float*)d_in[2];
  const float* W1 = (const float*)d_in[3];
  const float* W2 = (const float*)d_in[4];
  const float* b0 = (const float*)d_in[5];
  const float* b1 = (const float*)d_in[6];
  const float* b2 = (const float*)d_in[7];
  float* out = (float*)d_out;

  const int bs = in_sizes[0] / NV;        // 16384
  dim3 grid(bs / BT);                     // 1024 workgroups of 16 batches
  dim3 block(NWAVE * 32);                 // 10 waves
  nn_fused<<<grid, block, 0, stream>>>(x, M, W0, W1, W2, b0, b1, b2, out);
}